// ODEModel_13091060318537
// MI455X (gfx1250) — compile-verified
//
#include <hip/hip_runtime.h>
#include <math.h>

typedef __attribute__((ext_vector_type(2))) float v2f;
typedef __attribute__((ext_vector_type(8))) float v8f;

#define NSTEP 8192

__device__ __forceinline__ float fast_rcp(float x) {
#if __has_builtin(__builtin_amdgcn_rcpf)
    return __builtin_amdgcn_rcpf(x);     // v_rcp_f32, ~1 ulp
#else
    return 1.0f / x;
#endif
}

__device__ __forceinline__ float fast_tanh(float x) {
#if __has_builtin(__builtin_amdgcn_tanhf)
    return __builtin_amdgcn_tanhf(x);    // v_tanh_f32 (CDNA5 trans op)
#elif __has_builtin(__builtin_amdgcn_tanh_f32)
    return __builtin_amdgcn_tanh_f32(x);
#else
    // 1 - 2/(e^{2x}+1): monotone, saturates correctly at +/-1
    float e = __expf(2.0f * x);
    return 1.0f - 2.0f * fast_rcp(e + 1.0f);
#endif
}

__device__ __forceinline__ float fast_sigmoid(float x) {
    return fast_rcp(1.0f + __expf(-x));
}

__global__ __launch_bounds__(32) void ode_rk4_wmma(
    const float* __restrict__ s_grid,
    const float* __restrict__ y0,
    const float* __restrict__ W1,
    const float* __restrict__ b1,
    const float* __restrict__ W2,
    const float* __restrict__ b2,
    const float* __restrict__ W3,
    const float* __restrict__ b3,
    const float* __restrict__ pc,
    float* __restrict__ out)
{
    const int L   = threadIdx.x;   // 0..31, one wave32
    const int n16 = L & 15;        // N index within 16-wide WMMA tile
    const int hi  = L >> 4;        // 0: lanes 0-15 carry K={0,1}; 1: lanes 16-31 carry K={2,3}

    __shared__ float sh1[32];
    __shared__ float sh2[32];

    // ---- preload all weights into registers (once) ----
    float w1r[5];
#pragma unroll
    for (int j = 0; j < 5; ++j) w1r[j] = W1[L * 5 + j];
    const float b1v = b1[L];
    const float b2v = b2[L];
    const float b3v = (n16 < 4) ? b3[n16] : 0.0f;

    // B operands for V_WMMA_F32_16X16X4_F32: B[k][n] = Wt[n][4c+k]
    // lane layout (4x16 KxN, 2 VGPRs): lanes 0-15 hold K={0,1} in {v0,v1}, lanes 16-31 hold K={2,3}
    v2f B20[8], B21[8], B3c[8];
#pragma unroll
    for (int c = 0; c < 8; ++c) {
        const int kk = 4 * c + 2 * hi;
        B20[c].x = W2[(n16     ) * 32 + kk];
        B20[c].y = W2[(n16     ) * 32 + kk + 1];
        B21[c].x = W2[(n16 + 16) * 32 + kk];
        B21[c].y = W2[(n16 + 16) * 32 + kk + 1];
        if (n16 < 4) {
            B3c[c].x = W3[n16 * 32 + kk];
            B3c[c].y = W3[n16 * 32 + kk + 1];
        } else {
            B3c[c].x = 0.0f;
            B3c[c].y = 0.0f;
        }
    }

    float P[15];
#pragma unroll
    for (int i = 0; i < 15; ++i) P[i] = pc[i];

    // state (replicated across all lanes)
    float y[4];
#pragma unroll
    for (int j = 0; j < 4; ++j) y[j] = y0[j];

    if (L < 4) out[L] = y[L];   // ys[0] = y0

    // f(y, s) -> dy ; everything uniform across the wave, EXEC all ones at WMMAs
    auto evalf = [&](const float yy[4], float s, float dy[4]) {
        // layer 1: per-lane row of W1 (32x5)
        float h1p = b1v;
        h1p = fmaf(w1r[0], yy[0], h1p);
        h1p = fmaf(w1r[1], yy[1], h1p);
        h1p = fmaf(w1r[2], yy[2], h1p);
        h1p = fmaf(w1r[3], yy[3], h1p);
        h1p = fmaf(w1r[4], s,     h1p);
        const float h1 = fast_tanh(h1p);

        sh1[L] = h1;
        __syncthreads();   // single-wave workgroup: compiles to wave-level ordering only

        // layer 2: (32x32)@h1 via 2 chains of 8 x V_WMMA_F32_16X16X4_F32.
        // A = h1 chunk replicated over M; D[m][n] = out[n] for every m.
        v8f acc0 = {0.f,0.f,0.f,0.f,0.f,0.f,0.f,0.f};
        v8f acc1 = {0.f,0.f,0.f,0.f,0.f,0.f,0.f,0.f};
#pragma unroll
        for (int c = 0; c < 8; ++c) {
            v2f a;
            a.x = sh1[4 * c + 2 * hi];
            a.y = sh1[4 * c + 2 * hi + 1];
            acc0 = __builtin_amdgcn_wmma_f32_16x16x4_f32(false, a, false, B20[c],
                                                         (short)0, acc0, false, false);
            acc1 = __builtin_amdgcn_wmma_f32_16x16x4_f32(false, a, false, B21[c],
                                                         (short)0, acc1, false, false);
        }
        // lane L < 16 wants row L (chain0); lane L >= 16 wants row L (chain1)
        const float h2p = (hi ? acc1[0] : acc0[0]) + b2v;
        const float h2 = fast_tanh(h2p);

        sh2[L] = h2;
        __syncthreads();

        // layer 3: (4x32)@h2, one WMMA chain; columns n>=4 of B are zero
        v8f acc3 = {0.f,0.f,0.f,0.f,0.f,0.f,0.f,0.f};
#pragma unroll
        for (int c = 0; c < 8; ++c) {
            v2f a;
            a.x = sh2[4 * c + 2 * hi];
            a.y = sh2[4 * c + 2 * hi + 1];
            acc3 = __builtin_amdgcn_wmma_f32_16x16x4_f32(false, a, false, B3c[c],
                                                         (short)0, acc3, false, false);
        }
        const float sig = fast_sigmoid(acc3[0] + b3v);   // valid in lanes with n16<4
        const float n0 = __shfl(sig, 0, 32);
        const float n1 = __shfl(sig, 1, 32);
        const float n2 = __shfl(sig, 2, 32);
        const float n3 = __shfl(sig, 3, 32);

        const float ya = yy[0], yb = yy[1], yc = yy[2], yd = yy[3];
        dy[0] = fmaf(fmaf(P[0],  ya, P[1]),  ya, P[2])                        + n0;
        dy[1] = fmaf(fmaf(P[3],  yb, P[5]),  yb, fmaf(P[4],  ya * yb, P[6]))  + n1;
        dy[2] = fmaf(fmaf(P[7],  yc, P[9]),  yc, fmaf(P[8],  yb * yc, P[10])) + n2;
        dy[3] = fmaf(fmaf(P[11], yd, P[13]), yd, fmaf(P[12], yc * yd, P[14])) + n3;
    };

    float k1[4], k2[4], k3[4], k4[4], yt[4];
    for (int t = 0; t < NSTEP - 1; ++t) {
        const float s = s_grid[t];
        const float h = s_grid[t + 1] - s;

        evalf(y, s, k1);
#pragma unroll
        for (int j = 0; j < 4; ++j) yt[j] = fmaf(0.5f * h, k1[j], y[j]);
        evalf(yt, s, k2);
#pragma unroll
        for (int j = 0; j < 4; ++j) yt[j] = fmaf(0.5f * h, k2[j], y[j]);
        evalf(yt, s, k3);
#pragma unroll
        for (int j = 0; j < 4; ++j) yt[j] = fmaf(h, k3[j], y[j]);
        evalf(yt, s, k4);

        const float h6 = h * (1.0f / 6.0f);
#pragma unroll
        for (int j = 0; j < 4; ++j)
            y[j] = fmaf(h6, k1[j] + 2.0f * k2[j] + 2.0f * k3[j] + k4[j], y[j]);

        if (L < 4) out[(t + 1) * 4 + L] = y[L];
    }
}

extern "C" void kernel_launch(void* const* d_in, const int* in_sizes, int n_in,
                              void* d_out, int out_size, void* d_ws, size_t ws_size,
                              hipStream_t stream) {
    (void)in_sizes; (void)n_in; (void)out_size; (void)d_ws; (void)ws_size;
    const float* s_grid = (const float*)d_in[0];
    const float* y0     = (const float*)d_in[1];
    const float* W1     = (const float*)d_in[2];
    const float* b1     = (const float*)d_in[3];
    const float* W2     = (const float*)d_in[4];
    const float* b2     = (const float*)d_in[5];
    const float* W3     = (const float*)d_in[6];
    const float* b3     = (const float*)d_in[7];
    const float* pc     = (const float*)d_in[8];
    float* out          = (float*)d_out;

    // inherently serial scan: one wave32 owns the whole trajectory
    ode_rk4_wmma<<<1, 32, 0, stream>>>(s_grid, y0, W1, b1, W2, b2, W3, b3, pc, out);
}